// MTCNN_69140383531265
// MI455X (gfx1250) — compile-verified
//
#include <hip/hip_runtime.h>
#include <math.h>

// ---------------------------------------------------------------------------
// MTCNN stage-1 for MI455X (gfx1250, wave32).
// Conv layers: implicit GEMM on V_WMMA_F32_16X16X4_F32 (full fp32 WMMA).
//   wave tile = 16 output pixels (M) x 16 output channels (N), K = Cin*9.
//   Cin/Cout are template constants: the K-loop fully unrolls, the
//   k -> (ci,ky,kx) decomposition constant-folds, and all global loads are
//   UNconditional (addresses clamped in-bounds instead of lane-predicated,
//   since garbage A-rows / B-columns only affect outputs we never store).
//   Epilogue: bias/alpha hoisted as clamped loads -> branchless PReLU.
// Problem is tiny (~10 GFLOP, intermediates fit in 192MB L2) so fp32 WMMA is
// the right precision path; the tail (top-k/NMS) is LDS bitonic sort + serial
// greedy NMS, deterministic for graph replay.
// ---------------------------------------------------------------------------

#define B_IMG 4
#define H_IMG 720
#define W_IMG 1280
#define THRESH0 0.6f
#define OFFBOX (4.0f * 1280.0f)   // 4 * max(h, w)

typedef float v2f __attribute__((ext_vector_type(2)));
typedef float v8f __attribute__((ext_vector_type(8)));

// ------------------------------ resize -------------------------------------
// Area resize (box average, matching pooling-matrix einsum) + normalize.
__global__ void k_resize(const float* __restrict__ img, float* __restrict__ out,
                         int oh, int ow) {
  int i = blockIdx.x * blockDim.x + threadIdx.x;
  int total = B_IMG * 3 * oh * ow;
  if (i >= total) return;
  int ox = i % ow;
  int t  = i / ow;
  int oy = t % oh;
  int bc = t / oh;
  int sy = (oy * H_IMG) / oh;
  int ey = ((oy + 1) * H_IMG + oh - 1) / oh;   // ceil
  int sx = (ox * W_IMG) / ow;
  int ex = ((ox + 1) * W_IMG + ow - 1) / ow;
  const float* p = img + (size_t)bc * H_IMG * W_IMG;
  float s = 0.0f;
  for (int y = sy; y < ey; ++y)
    for (int x = sx; x < ex; ++x)
      s += p[y * W_IMG + x];
  float v = s / (float)((ey - sy) * (ex - sx));
  out[i] = (v - 127.5f) * 0.0078125f;
}

// --------------------------- WMMA 3x3 conv ---------------------------------
// A-element for compile-time k (after unroll): constant (ci,ky,kx).
// Unconditional load; caller guarantees arow is a valid in-bounds pixel.
__device__ __forceinline__ float aval(const float* __restrict__ arow,
                                      int HinWin, int Win, int k) {
  int ci = k / 9, r = k % 9;          // folds to literals after full unroll
  int ky = r / 3, kx = r % 3;
  return arow[(size_t)ci * HinWin + ky * Win + kx];
}

// D = A(16 px x K) * B(K x 16 ch) + bias + PReLU. One wave per 16-pixel x
// 16-channel tile; blockDim.x = 128 (4 waves). EXEC all-ones at WMMA (only
// wave-uniform early exit; OOB handled by clamped addresses, stores guarded).
template <int Cin, int Cout>
__global__ void k_conv3x3_wmma(const float* __restrict__ in,
                               const float* __restrict__ w,
                               const float* __restrict__ bias,
                               const float* __restrict__ alpha,
                               float* __restrict__ out,
                               int Hin, int Win, int Hout, int Wout) {
  constexpr int K  = Cin * 9;
  constexpr int KP = (K + 3) & ~3;
  const int lane = threadIdx.x & 31;
  const int wid  = threadIdx.x >> 5;
  const int half = lane >> 4;     // 0: lanes 0-15, 1: lanes 16-31
  const int l16  = lane & 15;
  const int HW   = Hout * Wout;
  const int tile = blockIdx.x * (blockDim.x >> 5) + wid;
  const int b    = blockIdx.y;
  const int nt   = blockIdx.z;
  const int pbase = tile * 16;
  if (pbase >= HW) return;        // wave-uniform: EXEC stays all-ones
  const int HinWin = Hin * Win;

  // Per-lane A-row position, computed ONCE; clamped so loads are always legal
  // (garbage rows are never stored -> no lane-predicated loads needed).
  int p  = pbase + l16;
  int pc = p < HW ? p : HW - 1;
  int py = pc / Wout, px = pc % Wout;
  const float* arow = in + ((size_t)b * Cin * Hin + py) * Win + px;

  // Per-lane B column (output channel), clamped the same way.
  const int n    = nt * 16 + l16;
  const bool nok = (n < Cout);
  const int nc   = nok ? n : Cout - 1;
  const float* wrow = w + (size_t)nc * K;

  __builtin_prefetch(arow, 0, 1);

  v8f acc = {};
#pragma unroll
  for (int k0 = 0; k0 < KP; k0 += 4) {
    // Four constant-k unconditional loads; lanes 0-15 consume k0..k0+1,
    // lanes 16-31 consume k0+2..k0+3 (V_WMMA_F32_16X16X4_F32 A/B layout).
    // (k0+i < K) is compile-time, so K-padding zeros fold away except in the
    // last step; pad elements are zero on BOTH operands -> contribute 0.
    float a0 = (k0 + 0 < K) ? aval(arow, HinWin, Win, k0 + 0) : 0.0f;
    float a1 = (k0 + 1 < K) ? aval(arow, HinWin, Win, k0 + 1) : 0.0f;
    float a2 = (k0 + 2 < K) ? aval(arow, HinWin, Win, k0 + 2) : 0.0f;
    float a3 = (k0 + 3 < K) ? aval(arow, HinWin, Win, k0 + 3) : 0.0f;
    float w0 = (k0 + 0 < K) ? wrow[k0 + 0] : 0.0f;
    float w1 = (k0 + 1 < K) ? wrow[k0 + 1] : 0.0f;
    float w2 = (k0 + 2 < K) ? wrow[k0 + 2] : 0.0f;
    float w3 = (k0 + 3 < K) ? wrow[k0 + 3] : 0.0f;
    v2f av, bv;
    av.x = half ? a2 : a0;
    av.y = half ? a3 : a1;
    bv.x = half ? w2 : w0;
    bv.y = half ? w3 : w1;
    acc = __builtin_amdgcn_wmma_f32_16x16x4_f32(false, av, false, bv,
                                                (short)0, acc, false, false);
  }

  // Hoisted clamped loads -> branchless PReLU in the store loop.
  const float bn = bias[nc];
  const float an = alpha[nc];
  float* orow = out + ((size_t)b * Cout + nc) * HW;

#pragma unroll
  for (int r = 0; r < 8; ++r) {
    int mm = r + half * 8;        // C/D layout: VGPR r -> M=r / M=r+8
    int pp = pbase + mm;
    float v = acc[r] + bn;
    v = v > 0.0f ? v : an * v;    // PReLU (v_cndmask + v_mul, no branch)
    if (pp < HW && nok) orow[pp] = v;
  }
}

// ------------------------------ maxpool ------------------------------------
__global__ void k_maxpool2(const float* __restrict__ in, float* __restrict__ out,
                           int BC, int H, int W, int Ho, int Wo) {
  int i = blockIdx.x * blockDim.x + threadIdx.x;
  int total = BC * Ho * Wo;
  if (i >= total) return;
  int x = i % Wo;
  int t = i / Wo;
  int y = t % Ho;
  int bc = t / Ho;
  const float* p = in + (size_t)bc * H * W;
  float m = -INFINITY;
#pragma unroll
  for (int dy = 0; dy < 2; ++dy)
#pragma unroll
    for (int dx = 0; dx < 2; ++dx) {
      int yy = 2 * y + dy, xx = 2 * x + dx;
      if (yy < H && xx < W) m = fmaxf(m, p[yy * W + xx]);
    }
  out[i] = m;
}

// ------------------------------ 1x1 heads ----------------------------------
__global__ void k_head(const float* __restrict__ in,
                       const float* __restrict__ w1, const float* __restrict__ b1,
                       const float* __restrict__ w2, const float* __restrict__ b2,
                       float* __restrict__ score, float* __restrict__ reg,
                       int HW) {
  int i = blockIdx.x * blockDim.x + threadIdx.x;
  int total = B_IMG * HW;
  if (i >= total) return;
  int b = i / HW, p = i % HW;
  const float* x = in + (size_t)b * 32 * HW + p;
  float l0 = b1[0], l1 = b1[1];
  float r0 = b2[0], r1 = b2[1], r2 = b2[2], r3 = b2[3];
#pragma unroll
  for (int c = 0; c < 32; ++c) {
    float v = x[(size_t)c * HW];
    l0 += w1[c] * v;       l1 += w1[32 + c] * v;
    r0 += w2[c] * v;       r1 += w2[32 + c] * v;
    r2 += w2[64 + c] * v;  r3 += w2[96 + c] * v;
  }
  float mx = fmaxf(l0, l1);
  float e0 = expf(l0 - mx), e1 = expf(l1 - mx);
  score[i] = e1 / (e0 + e1);
  reg[((size_t)b * 4 + 0) * HW + p] = r0;
  reg[((size_t)b * 4 + 1) * HW + p] = r1;
  reg[((size_t)b * 4 + 2) * HW + p] = r2;
  reg[((size_t)b * 4 + 3) * HW + p] = r3;
}

// ----------------------------- top-k machinery -----------------------------
__global__ void k_fill_keys(const float* __restrict__ score,
                            float* __restrict__ keys, int* __restrict__ idxs,
                            int N, int Npad) {
  int i = blockIdx.x * blockDim.x + threadIdx.x;
  if (i >= Npad) return;
  float k = -INFINITY;
  if (i < N) {
    float s = score[i];
    if (s >= THRESH0) k = s;
  }
  keys[i] = k;
  idxs[i] = i;
}

__global__ void k_fill_gkeys(const float* __restrict__ cb9,
                             const int* __restrict__ ckept,
                             float* __restrict__ keys, int* __restrict__ idxs,
                             int total, int pad) {
  int i = blockIdx.x * blockDim.x + threadIdx.x;
  if (i >= pad) return;
  float k = -INFINITY;
  if (i < total && ckept[i]) k = cb9[(size_t)i * 9 + 4];
  keys[i] = k;
  idxs[i] = i;
}

// Bitonic sort of one run (R <= 2048, power of two), descending.
__global__ void k_sort_runs(float* __restrict__ keys, int* __restrict__ idxs,
                            int R) {
  __shared__ float sk[2048];
  __shared__ int   si[2048];
  size_t base = (size_t)blockIdx.x * R;
  for (int t = threadIdx.x; t < R; t += blockDim.x) {
    sk[t] = keys[base + t];
    si[t] = idxs[base + t];
  }
  __syncthreads();
  for (int k = 2; k <= R; k <<= 1) {
    for (int j = k >> 1; j >= 1; j >>= 1) {
      for (int t = threadIdx.x; t < R; t += blockDim.x) {
        int l = t ^ j;
        if (l > t) {
          float a = sk[t], b = sk[l];
          bool ascBlk = ((t & k) != 0);
          bool sw = ascBlk ? (a > b) : (a < b);   // overall descending
          if (sw) {
            sk[t] = b; sk[l] = a;
            int ti = si[t]; si[t] = si[l]; si[l] = ti;
          }
        }
      }
      __syncthreads();
    }
  }
  for (int t = threadIdx.x; t < R; t += blockDim.x) {
    keys[base + t] = sk[t];
    idxs[base + t] = si[t];
  }
}

// Merge two sorted-descending runs, keep top R (bitonic merge in LDS).
__global__ void k_merge_runs(const float* __restrict__ ksrc,
                             const int* __restrict__ isrc,
                             float* __restrict__ kdst, int* __restrict__ idst,
                             int nruns, int R) {
  __shared__ float sk[4096];
  __shared__ int   si[4096];
  int ra = 2 * blockIdx.x, rb = ra + 1;
  for (int t = threadIdx.x; t < R; t += blockDim.x) {
    sk[t] = ksrc[(size_t)ra * R + t];
    si[t] = isrc[(size_t)ra * R + t];
  }
  if (rb < nruns) {
    for (int t = threadIdx.x; t < R; t += blockDim.x) {
      sk[2 * R - 1 - t] = ksrc[(size_t)rb * R + t];   // reversed -> bitonic
      si[2 * R - 1 - t] = isrc[(size_t)rb * R + t];
    }
  } else {
    for (int t = threadIdx.x; t < R; t += blockDim.x) {
      sk[R + t] = -INFINITY;
      si[R + t] = 0;
    }
  }
  __syncthreads();
  for (int j = R; j >= 1; j >>= 1) {
    for (int t = threadIdx.x; t < 2 * R; t += blockDim.x) {
      int l = t ^ j;
      if (l > t && sk[t] < sk[l]) {
        float a = sk[t]; sk[t] = sk[l]; sk[l] = a;
        int ti = si[t]; si[t] = si[l]; si[l] = ti;
      }
    }
    __syncthreads();
  }
  for (int t = threadIdx.x; t < R; t += blockDim.x) {
    kdst[(size_t)blockIdx.x * R + t] = sk[t];
    idst[(size_t)blockIdx.x * R + t] = si[t];
  }
}

// ---------------------------- candidate gather -----------------------------
__global__ void k_gather_scale(const float* __restrict__ keys,
                               const int* __restrict__ idxs,
                               const float* __restrict__ score,
                               const float* __restrict__ reg,
                               float* __restrict__ cb9, int* __restrict__ ci,
                               int* __restrict__ cvalid,
                               int kcount, int Hc, int Wc, float sc) {
  int t = blockIdx.x * blockDim.x + threadIdx.x;
  if (t >= kcount) return;
  float key = keys[t];
  int idx = idxs[t];
  int valid = (key > -1e30f) ? 1 : 0;
  if (!valid) idx = 0;
  int HW = Hc * Wc;
  int b = idx / HW, p = idx % HW;
  int cy = p / Wc, cx = p % Wc;
  float x1 = floorf((2.0f * cx + 1.0f) / sc);
  float y1 = floorf((2.0f * cy + 1.0f) / sc);
  float x2 = floorf((2.0f * cx + 12.0f) / sc);
  float y2 = floorf((2.0f * cy + 12.0f) / sc);
  cb9[(size_t)t * 9 + 0] = x1;
  cb9[(size_t)t * 9 + 1] = y1;
  cb9[(size_t)t * 9 + 2] = x2;
  cb9[(size_t)t * 9 + 3] = y2;
  cb9[(size_t)t * 9 + 4] = score[idx];
#pragma unroll
  for (int r = 0; r < 4; ++r)
    cb9[(size_t)t * 9 + 5 + r] = reg[((size_t)b * 4 + r) * HW + p];
  ci[t] = b;
  cvalid[t] = valid;
}

__global__ void k_gather_global(const float* __restrict__ keys,
                                const int* __restrict__ idxs,
                                const float* __restrict__ cb9,
                                const int* __restrict__ ci,
                                float* __restrict__ gb9, int* __restrict__ gii,
                                int* __restrict__ gvalid, int kg, int total) {
  int t = blockIdx.x * blockDim.x + threadIdx.x;
  if (t >= kg) return;
  float key = keys[t];
  int idx = idxs[t];
  int valid = (key > -1e30f) ? 1 : 0;
  if (!valid || idx >= total) idx = 0;
#pragma unroll
  for (int j = 0; j < 9; ++j)
    gb9[(size_t)t * 9 + j] = cb9[(size_t)idx * 9 + j];
  gii[t] = ci[idx];
  gvalid[t] = valid;
}

// ------------------------------- greedy NMS --------------------------------
// Inputs already sorted descending by masked score. Single block, serial scan.
__global__ void k_nms(const float* __restrict__ b9, const int* __restrict__ inds,
                      const int* __restrict__ valid_in, int* __restrict__ kept_out,
                      int n, float thresh, float off) {
  __shared__ float X1[2048], Y1[2048], X2[2048], Y2[2048], AR[2048];
  __shared__ unsigned char SUP[2048], VAL[2048], KEPT[2048];
  for (int t = threadIdx.x; t < n; t += blockDim.x) {
    float o = off * (float)inds[t];
    float x1 = b9[(size_t)t * 9 + 0] + o;
    float y1 = b9[(size_t)t * 9 + 1] + o;
    float x2 = b9[(size_t)t * 9 + 2] + o;
    float y2 = b9[(size_t)t * 9 + 3] + o;
    X1[t] = x1; Y1[t] = y1; X2[t] = x2; Y2[t] = y2;
    AR[t] = (x2 - x1 + 1.0f) * (y2 - y1 + 1.0f);
    SUP[t] = 0;
    VAL[t] = (unsigned char)valid_in[t];
    KEPT[t] = 0;
  }
  __syncthreads();
  for (int i = 0; i < n; ++i) {
    bool kept_i = VAL[i] && !SUP[i];
    if (threadIdx.x == 0) KEPT[i] = kept_i ? 1 : 0;
    if (kept_i) {
      for (int j = i + 1 + threadIdx.x; j < n; j += blockDim.x) {
        float iw = fminf(X2[i], X2[j]) - fmaxf(X1[i], X1[j]) + 1.0f;
        float ih = fminf(Y2[i], Y2[j]) - fmaxf(Y1[i], Y1[j]) + 1.0f;
        iw = fmaxf(iw, 0.0f);
        ih = fmaxf(ih, 0.0f);
        float inter = iw * ih;
        float iou = inter / (AR[i] + AR[j] - inter);
        if (iou > thresh) SUP[j] = 1;
      }
    }
    __syncthreads();
  }
  for (int t = threadIdx.x; t < n; t += blockDim.x) kept_out[t] = KEPT[t];
}

// ------------------------------ refinement ---------------------------------
__global__ void k_refine(const float* __restrict__ gb9,
                         const int* __restrict__ gkept,
                         const int* __restrict__ gii,
                         float* __restrict__ out, int kg) {
  int t = blockIdx.x * blockDim.x + threadIdx.x;
  if (t >= kg) return;
  int valid = gkept[t];
  float x1 = gb9[(size_t)t * 9 + 0], y1 = gb9[(size_t)t * 9 + 1];
  float x2 = gb9[(size_t)t * 9 + 2], y2 = gb9[(size_t)t * 9 + 3];
  float sc = gb9[(size_t)t * 9 + 4];
  float r0 = gb9[(size_t)t * 9 + 5], r1 = gb9[(size_t)t * 9 + 6];
  float r2 = gb9[(size_t)t * 9 + 7], r3 = gb9[(size_t)t * 9 + 8];
  float regw = x2 - x1, regh = y2 - y1;
  float qq1 = x1 + r0 * regw, qq2 = y1 + r1 * regh;
  float qq3 = x2 + r2 * regw, qq4 = y2 + r3 * regh;
  float bw = qq3 - qq1, bh = qq4 - qq2;
  float l = fmaxf(bw, bh);
  float x1n = qq1 + bw * 0.5f - l * 0.5f;
  float y1n = qq2 + bh * 0.5f - l * 0.5f;
  float b5[5] = {x1n, y1n, x1n + l, y1n + l, sc};
#pragma unroll
  for (int j = 0; j < 5; ++j)
    out[(size_t)t * 5 + j] = valid ? b5[j] : 0.0f;
  out[(size_t)kg * 5 + t] = valid ? 1.0f : 0.0f;  // valid mask
  out[(size_t)kg * 6 + t] = (float)gii[t];        // batch index
}

// ------------------------------- host side ---------------------------------
extern "C" void kernel_launch(void* const* d_in, const int* in_sizes, int n_in,
                              void* d_out, int out_size, void* d_ws, size_t ws_size,
                              hipStream_t stream) {
  (void)in_sizes; (void)n_in; (void)out_size; (void)ws_size;
  const float* imgs = (const float*)d_in[0];
  const float* c1w  = (const float*)d_in[1];
  const float* c1b  = (const float*)d_in[2];
  const float* a1   = (const float*)d_in[3];
  const float* c2w  = (const float*)d_in[4];
  const float* c2b  = (const float*)d_in[5];
  const float* a2   = (const float*)d_in[6];
  const float* c3w  = (const float*)d_in[7];
  const float* c3b  = (const float*)d_in[8];
  const float* a3   = (const float*)d_in[9];
  const float* c41w = (const float*)d_in[10];
  const float* c41b = (const float*)d_in[11];
  const float* c42w = (const float*)d_in[12];
  const float* c42b = (const float*)d_in[13];

  // ---- workspace carve (bump allocator, 256B aligned) ----
  char* wsb = (char*)d_ws;
  size_t cur = 0;
  auto carve = [&](size_t bytes) -> void* {
    void* p = wsb + cur;
    cur += (bytes + 255) & ~(size_t)255;
    return p;
  };
  float* IMG   = (float*)carve((size_t)4100000 * 4);    // resized image
  float* TA    = (float*)carve((size_t)13400000 * 4);   // ping
  float* TB    = (float*)carve((size_t)10500000 * 4);   // pong
  float* SCORE = (float*)carve((size_t)340000 * 4);
  float* REGB  = (float*)carve((size_t)1360000 * 4);
  float* KEYA  = (float*)carve((size_t)340000 * 4);
  int*   IDXA  = (int*)  carve((size_t)340000 * 4);
  float* KEYB  = (float*)carve((size_t)340000 * 4);
  int*   IDXB  = (int*)  carve((size_t)340000 * 4);
  float* CB9   = (float*)carve((size_t)16384 * 9 * 4);
  int*   CI    = (int*)  carve((size_t)16384 * 4);
  int*   CVAL  = (int*)  carve((size_t)16384 * 4);
  int*   CKEPT = (int*)  carve((size_t)16384 * 4);
  float* GKA   = (float*)carve((size_t)16384 * 4);
  int*   GIA   = (int*)  carve((size_t)16384 * 4);
  float* GKB   = (float*)carve((size_t)16384 * 4);
  int*   GIB   = (int*)  carve((size_t)16384 * 4);
  float* GB9   = (float*)carve((size_t)2048 * 9 * 4);
  int*   GII   = (int*)  carve((size_t)2048 * 4);
  int*   GVAL  = (int*)  carve((size_t)2048 * 4);
  int*   GKEPT = (int*)  carve((size_t)2048 * 4);

  // ---- scale pyramid (host, deterministic) ----
  double scales[16];
  int nscales = 0;
  {
    double m = 12.0 / 20.0;
    double minl = 720.0 * m;  // min(H,W) * m
    double s = m;
    while (minl >= 12.0 && nscales < 16) {
      scales[nscales++] = s;
      s *= 0.709;
      minl *= 0.709;
    }
  }

  int candOff = 0;
  for (int si = 0; si < nscales; ++si) {
    double sc = scales[si];
    int oh = (int)((double)H_IMG * sc + 1.0);
    int ow = (int)((double)W_IMG * sc + 1.0);
    int h1 = oh - 2, w1 = ow - 2;             // conv1 out
    int hp = (h1 + 1) / 2, wp = (w1 + 1) / 2; // maxpool (ceil)
    int h2 = hp - 2, w2 = wp - 2;             // conv2 out
    int Hc = h2 - 2, Wc = w2 - 2;             // conv3 / head out
    if (Hc <= 0 || Wc <= 0) continue;

    {   // resize + normalize
      int total = B_IMG * 3 * oh * ow;
      k_resize<<<(total + 255) / 256, 256, 0, stream>>>(imgs, IMG, oh, ow);
    }
    {   // conv1 3->10 + PReLU (WMMA)
      dim3 g((h1 * w1 + 63) / 64, B_IMG, 1);
      k_conv3x3_wmma<3, 10><<<g, 128, 0, stream>>>(IMG, c1w, c1b, a1, TA,
                                                   oh, ow, h1, w1);
    }
    {   // maxpool 2x2 ceil
      int total = B_IMG * 10 * hp * wp;
      k_maxpool2<<<(total + 255) / 256, 256, 0, stream>>>(TA, TB, B_IMG * 10,
                                                          h1, w1, hp, wp);
    }
    {   // conv2 10->16 + PReLU (WMMA)
      dim3 g((h2 * w2 + 63) / 64, B_IMG, 1);
      k_conv3x3_wmma<10, 16><<<g, 128, 0, stream>>>(TB, c2w, c2b, a2, TA,
                                                    hp, wp, h2, w2);
    }
    {   // conv3 16->32 + PReLU (WMMA, two N tiles)
      dim3 g((Hc * Wc + 63) / 64, B_IMG, 2);
      k_conv3x3_wmma<16, 32><<<g, 128, 0, stream>>>(TA, c3w, c3b, a3, TB,
                                                    h2, w2, Hc, Wc);
    }
    {   // 1x1 heads + softmax
      int HW = Hc * Wc;
      int total = B_IMG * HW;
      k_head<<<(total + 255) / 256, 256, 0, stream>>>(TB, c41w, c41b,
                                                      c42w, c42b, SCORE, REGB, HW);
    }

    // per-scale top-k (sort 1024-runs, merge tree)
    int N = B_IMG * Hc * Wc;
    int nruns = (N + 1023) / 1024;
    int Npad = nruns * 1024;
    k_fill_keys<<<(Npad + 255) / 256, 256, 0, stream>>>(SCORE, KEYA, IDXA, N, Npad);
    k_sort_runs<<<nruns, 256, 0, stream>>>(KEYA, IDXA, 1024);
    float* skeys = KEYA; int* sidx = IDXA;
    float* dkeys = KEYB; int* didx = IDXB;
    int nr = nruns;
    while (nr > 1) {
      int onr = (nr + 1) / 2;
      k_merge_runs<<<onr, 256, 0, stream>>>(skeys, sidx, dkeys, didx, nr, 1024);
      float* tk = skeys; skeys = dkeys; dkeys = tk;
      int* ti = sidx; sidx = didx; didx = ti;
      nr = onr;
    }
    int kcount = N < 1024 ? N : 1024;
    k_gather_scale<<<(kcount + 255) / 256, 256, 0, stream>>>(
        skeys, sidx, SCORE, REGB, CB9 + (size_t)candOff * 9, CI + candOff,
        CVAL + candOff, kcount, Hc, Wc, (float)sc);
    k_nms<<<1, 256, 0, stream>>>(CB9 + (size_t)candOff * 9, CI + candOff,
                                 CVAL + candOff, CKEPT + candOff, kcount,
                                 0.5f, OFFBOX);
    candOff += kcount;
  }

  // ---- global top-2048 + NMS + refine ----
  int total = candOff;
  int pad = 2048;
  while (pad < total) pad <<= 1;
  int nrg = pad / 2048;
  k_fill_gkeys<<<(pad + 255) / 256, 256, 0, stream>>>(CB9, CKEPT, GKA, GIA,
                                                      total, pad);
  k_sort_runs<<<nrg, 256, 0, stream>>>(GKA, GIA, 2048);
  float* skeys = GKA; int* sidx = GIA;
  float* dkeys = GKB; int* didx = GIB;
  int nr = nrg;
  while (nr > 1) {
    int onr = (nr + 1) / 2;
    k_merge_runs<<<onr, 256, 0, stream>>>(skeys, sidx, dkeys, didx, nr, 2048);
    float* tk = skeys; skeys = dkeys; dkeys = tk;
    int* ti = sidx; sidx = didx; didx = ti;
    nr = onr;
  }
  int kg = total < 2048 ? total : 2048;
  k_gather_global<<<(kg + 255) / 256, 256, 0, stream>>>(skeys, sidx, CB9, CI,
                                                        GB9, GII, GVAL, kg, total);
  k_nms<<<1, 256, 0, stream>>>(GB9, GII, GVAL, GKEPT, kg, 0.7f, OFFBOX);
  k_refine<<<(kg + 255) / 256, 256, 0, stream>>>(GB9, GKEPT, GII,
                                                 (float*)d_out, kg);
}